// GCN_85899346455
// MI455X (gfx1250) — compile-verified
//
#include <hip/hip_runtime.h>
#include <hip/hip_bf16.h>

typedef float v2f __attribute__((ext_vector_type(2)));
typedef float v8f __attribute__((ext_vector_type(8)));

#define FEAT 512
#define HID  16

// ---------------------------------------------------------------------------
// small helpers
// ---------------------------------------------------------------------------
__global__ void init_one_kernel(float* __restrict__ deg, int n) {
    int i = blockIdx.x * blockDim.x + threadIdx.x;
    if (i < n) deg[i] = 1.0f;                 // self-loop contribution
}

__global__ void degree_kernel(const long long* __restrict__ cols,
                              float* __restrict__ deg, int E) {
    int i = blockIdx.x * blockDim.x + threadIdx.x;
    if (i < E) {
        int c = (int)cols[i];
        unsafeAtomicAdd(&deg[c], 1.0f);       // hw global_atomic_add_f32
    }
}

__global__ void rsqrt_kernel(float* __restrict__ deg, int n) {
    int i = blockIdx.x * blockDim.x + threadIdx.x;
    if (i < n) {
        float d = deg[i];
        deg[i] = (d > 0.0f) ? __frsqrt_rn(d) : 0.0f;   // becomes dinv in-place
    }
}

// ---------------------------------------------------------------------------
// GEMM1: h = x @ W1   (N x 512) * (512 x 16) -> (N x 16), fp32 WMMA
// one wave32 handles one 16-row tile; W1 staged in LDS (32 KB)
// ---------------------------------------------------------------------------
__global__ __launch_bounds__(128) void gemm1_wmma_kernel(
    const float* __restrict__ x, const float* __restrict__ W1,
    float* __restrict__ h, int n_nodes)
{
    __shared__ float lW[FEAT * HID];          // 32 KB
    const int tid = threadIdx.x;

    // cooperative load of W1 into LDS (float4, coalesced)
    {
        const float4* Wv  = (const float4*)W1;
        float4*       lWv = (float4*)lW;
        for (int i = tid; i < (FEAT * HID) / 4; i += blockDim.x) lWv[i] = Wv[i];
    }
    __syncthreads();

    const int wave = tid >> 5;
    const int lane = tid & 31;
    const int tile = blockIdx.x * 4 + wave;
    if (tile * 16 >= n_nodes) return;         // wave-uniform exit (EXEC all 1s for WMMA)

    const int m    = lane & 15;               // M (node within tile), N column for B
    const int half = lane >> 4;               // lane-half selects K pair
    const int koff = half * 2;

    const float* __restrict__ arow = x + (size_t)(tile * 16 + m) * FEAT;

    v8f c = {0.f, 0.f, 0.f, 0.f, 0.f, 0.f, 0.f, 0.f};

    for (int k0 = 0; k0 < FEAT; k0 += 32) {
        if (k0 + 32 < FEAT)
            __builtin_prefetch(arow + k0 + 32, 0, 1);   // global_prefetch_b8
#pragma unroll
        for (int kk = 0; kk < 32; kk += 4) {
            const int k = k0 + kk;
            v2f a, b;
            // A 16x4: lane<16 holds K=k..k+1, lane>=16 holds K=k+2..k+3 (row m)
            a.x = arow[k + koff];
            a.y = arow[k + koff + 1];
            // B 4x16: VGPR0 -> rows K=k (lo half) / k+2 (hi half), VGPR1 -> k+1 / k+3
            b.x = lW[(k + koff)     * HID + m];
            b.y = lW[(k + koff + 1) * HID + m];
            c = __builtin_amdgcn_wmma_f32_16x16x4_f32(
                    false, a, false, b, (short)0, c, false, false);
        }
    }

    // D 16x16: VGPR r -> M = r + half*8, N = m
    float* __restrict__ hout = h + (size_t)tile * 16 * HID;
#pragma unroll
    for (int r = 0; r < 8; ++r)
        hout[(r + half * 8) * HID + m] = c[r];
}

// ---------------------------------------------------------------------------
// GEMM2: h2 = relu(agg + b1) @ W2   (N x 16) * (16 x 16) -> (N x 16)
// bias + ReLU fused into the A-fragment load
// ---------------------------------------------------------------------------
__global__ __launch_bounds__(256) void gemm2_wmma_kernel(
    const float* __restrict__ agg, const float* __restrict__ W2,
    const float* __restrict__ b1, float* __restrict__ h2, int n_nodes)
{
    __shared__ float lW[HID * HID];
    __shared__ float lb[HID];
    const int tid = threadIdx.x;
    if (tid < HID * HID) lW[tid] = W2[tid];
    if (tid < HID)       lb[tid] = b1[tid];
    __syncthreads();

    const int wave = tid >> 5;
    const int lane = tid & 31;
    const int tile = blockIdx.x * 8 + wave;
    if (tile * 16 >= n_nodes) return;

    const int m    = lane & 15;
    const int half = lane >> 4;
    const int koff = half * 2;

    const float* __restrict__ arow = agg + (size_t)(tile * 16 + m) * HID;

    v8f c = {0.f, 0.f, 0.f, 0.f, 0.f, 0.f, 0.f, 0.f};
#pragma unroll
    for (int k0 = 0; k0 < HID; k0 += 4) {
        v2f a, b;
        a.x = fmaxf(arow[k0 + koff]     + lb[k0 + koff],     0.0f);
        a.y = fmaxf(arow[k0 + koff + 1] + lb[k0 + koff + 1], 0.0f);
        b.x = lW[(k0 + koff)     * HID + m];
        b.y = lW[(k0 + koff + 1) * HID + m];
        c = __builtin_amdgcn_wmma_f32_16x16x4_f32(
                false, a, false, b, (short)0, c, false, false);
    }

    float* __restrict__ hout = h2 + (size_t)tile * 16 * HID;
#pragma unroll
    for (int r = 0; r < 8; ++r)
        hout[(r + half * 8) * HID + m] = c[r];
}

// ---------------------------------------------------------------------------
// Edge scatter: agg[col] += h[row] * dinv[row]*dinv[col]
// 16 lanes per edge -> coalesced 64B gather + contiguous 16-float atomic burst
// edges [0,E): from edge_index; edges [E, E+N): self loops
// ---------------------------------------------------------------------------
__global__ void scatter_kernel(const float* __restrict__ h,
                               const float* __restrict__ dinv,
                               const long long* __restrict__ rows,
                               const long long* __restrict__ cols,
                               float* __restrict__ agg,
                               int E, int n_nodes)
{
    const long long gid = (long long)blockIdx.x * blockDim.x + threadIdx.x;
    const int  comp = (int)(gid & 15);
    const long long e = gid >> 4;
    const long long total = (long long)E + n_nodes;
    if (e >= total) return;

    int r, c;
    if (e < E) {
        r = (int)rows[e];                 // broadcast within the 16-lane group
        c = (int)cols[e];
    } else {
        r = c = (int)(e - (long long)E);  // self loop
    }
    const float nv  = dinv[r] * dinv[c];  // L0/L2-resident broadcast loads
    const float val = h[(size_t)r * HID + comp] * nv;
    unsafeAtomicAdd(&agg[(size_t)c * HID + comp], val);
}

// ---------------------------------------------------------------------------
// out[i] = relu(agg2[i] + b2) . Wl + bl
// ---------------------------------------------------------------------------
__global__ void final_kernel(const float* __restrict__ agg2,
                             const float* __restrict__ b2,
                             const float* __restrict__ Wl,
                             const float* __restrict__ bl,
                             float* __restrict__ out, int n)
{
    int i = blockIdx.x * blockDim.x + threadIdx.x;
    if (i >= n) return;
    const float4* a  = (const float4*)(agg2 + (size_t)i * HID);
    const float4* bb = (const float4*)b2;
    const float4* wv = (const float4*)Wl;
    float acc = 0.0f;
#pragma unroll
    for (int q = 0; q < 4; ++q) {
        float4 t = a[q], b = bb[q], w = wv[q];
        acc += fmaxf(t.x + b.x, 0.f) * w.x;
        acc += fmaxf(t.y + b.y, 0.f) * w.y;
        acc += fmaxf(t.z + b.z, 0.f) * w.z;
        acc += fmaxf(t.w + b.w, 0.f) * w.w;
    }
    out[i] = acc + bl[0];
}

// ---------------------------------------------------------------------------
extern "C" void kernel_launch(void* const* d_in, const int* in_sizes, int n_in,
                              void* d_out, int out_size, void* d_ws, size_t ws_size,
                              hipStream_t stream)
{
    const float*     x   = (const float*)d_in[0];
    const long long* ei  = (const long long*)d_in[1];  // int64 per reference
    const float*     W1  = (const float*)d_in[2];
    const float*     b1  = (const float*)d_in[3];
    const float*     W2  = (const float*)d_in[4];
    const float*     b2  = (const float*)d_in[5];
    const float*     Wl  = (const float*)d_in[6];
    const float*     bl  = (const float*)d_in[7];
    float*           out = (float*)d_out;

    const int n = in_sizes[0] / FEAT;   // 100000
    const int E = in_sizes[1] / 2;      // 3200000
    const long long* rows = ei;
    const long long* cols = ei + E;

    // workspace layout: dinv[n] | bufA[n*16] | bufB[n*16]  (~13.3 MB)
    char*  ws   = (char*)d_ws;
    float* dinv = (float*)ws;
    size_t off  = (((size_t)n * sizeof(float)) + 255) & ~(size_t)255;
    float* bufA = (float*)(ws + off);          // h1, later h2
    float* bufB = bufA + (size_t)n * HID;      // agg1, later agg2

    const size_t featBytes = (size_t)n * HID * sizeof(float);

    // 1) degree + rsqrt -> dinv
    init_one_kernel<<<(n + 255) / 256, 256, 0, stream>>>(dinv, n);
    degree_kernel<<<(E + 255) / 256, 256, 0, stream>>>(cols, dinv, E);
    rsqrt_kernel<<<(n + 255) / 256, 256, 0, stream>>>(dinv, n);

    // 2) h1 = x @ W1  (WMMA)
    const int tiles = (n + 15) / 16;
    gemm1_wmma_kernel<<<(tiles + 3) / 4, 128, 0, stream>>>(x, W1, bufA, n);

    // 3) agg1 = scatter(h1 * norm)
    hipMemsetAsync(bufB, 0, featBytes, stream);
    const long long sthreads = ((long long)E + n) * 16;
    const unsigned  sblocks  = (unsigned)((sthreads + 255) / 256);
    scatter_kernel<<<sblocks, 256, 0, stream>>>(bufA, dinv, rows, cols, bufB, E, n);

    // 4) h2 = relu(agg1 + b1) @ W2  (WMMA, fused bias+relu)
    gemm2_wmma_kernel<<<(tiles + 7) / 8, 256, 0, stream>>>(bufB, W2, b1, bufA, n);

    // 5) agg2 = scatter(h2 * norm)
    hipMemsetAsync(bufB, 0, featBytes, stream);
    scatter_kernel<<<sblocks, 256, 0, stream>>>(bufA, dinv, rows, cols, bufB, E, n);

    // 6) out = relu(agg2 + b2) . Wl + bl
    final_kernel<<<(n + 255) / 256, 256, 0, stream>>>(bufB, b2, Wl, bl, out, n);
}